// PhysicsNet_77859167142443
// MI455X (gfx1250) — compile-verified
//
#include <hip/hip_runtime.h>

typedef float v2f __attribute__((ext_vector_type(2)));
typedef float v8f __attribute__((ext_vector_type(8)));

#define N_IPM 30
#define SIGMA 0.1f

// Sum of 32 lane values via V_WMMA_F32_16X16X4_F32 with all-ones B.
// A[m,0]=c_m (lanes 0-15, vgpr0), A[m,2]=c_{16+m} (lanes 16-31, vgpr0), other K slots 0.
// D[m,n] = c_m + c_{16+m}. Lanes<16 sum rows 0-7 of one column, lanes>=16 rows 8-15;
// shfl_xor(16) combines -> every lane holds the full 32-lane sum.
// Requires EXEC == all ones (call only from fully-active waves).
__device__ __forceinline__ float wave_sum_wmma(float x) {
    v2f a; a.x = x;    a.y = 0.0f;
    v2f b; b.x = 1.0f; b.y = 1.0f;   // all-ones 4x16 B
    v8f c = {};
    c = __builtin_amdgcn_wmma_f32_16x16x4_f32(false, a, false, b, (short)0, c, false, false);
    float t = c[0] + c[1] + c[2] + c[3] + c[4] + c[5] + c[6] + c[7];
    t += __shfl_xor(t, 16, 32);
    return t;
}

__global__ __launch_bounds__(256) void ipm_kernel(const float* __restrict__ data,
                                                  const float* __restrict__ mu,
                                                  float* __restrict__ partials, int n) {
    const int i = blockIdx.x * blockDim.x + threadIdx.x;

    float vs = 0.f, vn = 0.f, us = 0.f;
    if (i < n) { vs = data[3 * i]; vn = data[3 * i + 1]; us = data[3 * i + 2]; }
    const float mu0  = mu[0];
    const float beta = vn - vs - us;
    const float w    = vs + us;

    // Constant-folded p and h (only nonzero slots kept).
    const float p0 = -20.f * beta + w;
    const float p1 =  20.f * beta - w;
    const float p2 = mu0;                 // p[3..8] = 100
    const float h6 = w, h7 = -w, h8 = mu0; // other h = 0

    float z[9], s[12], l[12];
#pragma unroll
    for (int k = 0; k < 9; ++k) z[k] = 0.f;
#pragma unroll
    for (int k = 0; k < 12; ++k) { s[k] = 1.f; l[k] = 1.f; }

#pragma unroll 1
    for (int it = 0; it < N_IPM; ++it) {
        // rp = R z + s - h   (R is sparse/constant)
        float rp[12];
#pragma unroll
        for (int k = 0; k < 6; ++k) rp[k] = s[k] - z[3 + k];
        rp[6] = -z[0] + z[1] - z[2] + s[6] - h6;
        rp[7] =  z[0] - z[1] - z[2] + s[7] - h7;
        rp[8] =  z[0] + z[1]        + s[8] - h8;
        rp[9]  = s[9]  - z[0];
        rp[10] = s[10] - z[1];
        rp[11] = s[11] - z[2];

        // rd = Qm z + p + R^T lam
        float rd[9];
        rd[0] = 22.001f * z[0] - 22.f * z[1] + p0 + (-l[6] + l[7] + l[8] - l[9]);
        rd[1] = -22.f * z[0] + 22.001f * z[1] + p1 + ( l[6] - l[7] + l[8] - l[10]);
        rd[2] = 0.001f * z[2] + p2 + (-l[6] - l[7] - l[11]);
#pragma unroll
        for (int a = 0; a < 6; ++a) rd[3 + a] = 0.001f * z[3 + a] + 100.f - l[a];

        // complementarity
        float dot = 0.f;
#pragma unroll
        for (int k = 0; k < 12; ++k) dot += l[k] * s[k];
        const float mubar = dot * (1.f / 12.f);

        float rc[12], sinv[12], W[12], g[12];
#pragma unroll
        for (int k = 0; k < 12; ++k) {
            sinv[k] = 1.f / s[k];
            rc[k]   = l[k] * s[k] - SIGMA * mubar;
            W[k]    = l[k] * sinv[k];
            g[k]    = (l[k] * rp[k] - rc[k]) * sinv[k];
        }

        // rhs = -(rd + g @ R)
        const float r0 = -(rd[0] + (-g[6] + g[7] + g[8] - g[9]));
        const float r1 = -(rd[1] + ( g[6] - g[7] + g[8] - g[10]));
        const float r2 = -(rd[2] + (-g[6] - g[7] - g[11]));
        float rz[6];
#pragma unroll
        for (int a = 0; a < 6; ++a) rz[a] = -(rd[3 + a] - g[a]);

        // M = Qm + R^T W R is block-diagonal: dense SPD 3x3 + diag(0.001 + W[0..5])
        const float m00 = 22.001f + W[6] + W[7] + W[8] + W[9];
        const float m01 = -22.f   - W[6] - W[7] + W[8];
        const float m02 =  W[6] - W[7];
        const float m11 = 22.001f + W[6] + W[7] + W[8] + W[10];
        const float m12 = -W[6] + W[7];
        const float m22 = 0.001f  + W[6] + W[7] + W[11];

        // 3x3 adjugate solve
        const float a00 = m11 * m22 - m12 * m12;
        const float a01 = m02 * m12 - m01 * m22;
        const float a02 = m01 * m12 - m02 * m11;
        const float a11 = m00 * m22 - m02 * m02;
        const float a12 = m01 * m02 - m00 * m12;
        const float a22 = m00 * m11 - m01 * m01;
        const float dinv = 1.f / (m00 * a00 + m01 * a01 + m02 * a02);

        float dz[9];
        dz[0] = (a00 * r0 + a01 * r1 + a02 * r2) * dinv;
        dz[1] = (a01 * r0 + a11 * r1 + a12 * r2) * dinv;
        dz[2] = (a02 * r0 + a12 * r1 + a22 * r2) * dinv;
#pragma unroll
        for (int a = 0; a < 6; ++a) dz[3 + a] = rz[a] / (0.001f + W[a]);

        // ds = -rp - R dz ; dlam = -(rc + lam*ds)/s
        float ds[12], dl[12];
#pragma unroll
        for (int k = 0; k < 6; ++k) ds[k] = -rp[k] + dz[3 + k];
        ds[6] = -rp[6] + dz[0] - dz[1] + dz[2];
        ds[7] = -rp[7] - dz[0] + dz[1] + dz[2];
        ds[8] = -rp[8] - dz[0] - dz[1];
        ds[9]  = -rp[9]  + dz[0];
        ds[10] = -rp[10] + dz[1];
        ds[11] = -rp[11] + dz[2];
#pragma unroll
        for (int k = 0; k < 12; ++k) dl[k] = -(rc[k] + l[k] * ds[k]) * sinv[k];

        // fraction-to-boundary step
        float am = 1e30f;
#pragma unroll
        for (int k = 0; k < 12; ++k) {
            float cs = (ds[k] < 0.f) ? (-s[k] / ds[k]) : 1e30f;
            float cl = (dl[k] < 0.f) ? (-l[k] / dl[k]) : 1e30f;
            am = fminf(am, fminf(cs, cl));
        }
        const float alpha = fminf(1.f, 0.99f * am);

#pragma unroll
        for (int k = 0; k < 9;  ++k) z[k] += alpha * dz[k];
#pragma unroll
        for (int k = 0; k < 12; ++k) { s[k] += alpha * ds[k]; l[k] += alpha * dl[k]; }
    }

    // cost = 0.5 z'Qm z + p.z + A1*beta^2
    float quad = 22.001f * (z[0] * z[0] + z[1] * z[1]) - 44.f * z[0] * z[1];
#pragma unroll
    for (int k = 2; k < 9; ++k) quad += 0.001f * z[k] * z[k];
    float pz = p0 * z[0] + p1 * z[1] + p2 * z[2];
#pragma unroll
    for (int k = 3; k < 9; ++k) pz += 100.f * z[k];
    float cost = 0.5f * quad + pz + 10.f * beta * beta;
    if (i >= n) cost = 0.f;

    // block reduction: WMMA wave-sum, then 8 wave partials via LDS
    __shared__ float wsum[8];
    const float t = wave_sum_wmma(cost);
    const int lane = threadIdx.x & 31;
    const int wid  = threadIdx.x >> 5;
    if (lane == 0) wsum[wid] = t;
    __syncthreads();
    if (threadIdx.x == 0) {
        float bsum = 0.f;
#pragma unroll
        for (int k = 0; k < 8; ++k) bsum += wsum[k];
        partials[blockIdx.x] = bsum;
    }
}

// Single-wave final reduction: accumulate 64 partials per WMMA through the C operand.
__global__ __launch_bounds__(32) void reduce_kernel(const float* __restrict__ partials,
                                                    int nparts, float* __restrict__ out) {
    const int lane = threadIdx.x;  // 0..31, one full wave -> EXEC all ones
    v2f b; b.x = 1.0f; b.y = 1.0f;
    v8f acc = {};
    for (int base = 0; base < nparts; base += 64) {
        v2f a;
        const int i0 = base + lane, i1 = base + 32 + lane;
        a.x = (i0 < nparts) ? partials[i0] : 0.f;
        a.y = (i1 < nparts) ? partials[i1] : 0.f;
        acc = __builtin_amdgcn_wmma_f32_16x16x4_f32(false, a, false, b, (short)0, acc, false, false);
    }
    float t = acc[0] + acc[1] + acc[2] + acc[3] + acc[4] + acc[5] + acc[6] + acc[7];
    t += __shfl_xor(t, 16, 32);
    if (lane == 0) out[0] = t;
}

extern "C" void kernel_launch(void* const* d_in, const int* in_sizes, int n_in,
                              void* d_out, int out_size, void* d_ws, size_t ws_size,
                              hipStream_t stream) {
    const float* data = (const float*)d_in[0];
    const float* mu   = (const float*)d_in[1];
    const int n = in_sizes[0] / 3;
    const int blocks = (n + 255) / 256;
    float* partials = (float*)d_ws;

    ipm_kernel<<<blocks, 256, 0, stream>>>(data, mu, partials, n);
    reduce_kernel<<<1, 32, 0, stream>>>(partials, blocks, (float*)d_out);
}